// GCNNet_16982300688850
// MI455X (gfx1250) — compile-verified
//
#include <hip/hip_runtime.h>

typedef float v2f __attribute__((ext_vector_type(2)));
typedef float v8f __attribute__((ext_vector_type(8)));

#define DF 300
#define K4 75            // DF / 4 (K-steps of 4)
#define NPAD 304         // padded row count for W (19 full 16-col tiles)
#define FRAG_STRIDE 66   // floats per K-step slab in LDS (conflict-free, 8B aligned)
#define PIPE 3           // software pipeline depth ((K4 - PIPE) % PIPE == 0)

#define WMMA_F32(a, b, c) \
    __builtin_amdgcn_wmma_f32_16x16x4_f32(false, (a), false, (b), (short)0, (c), false, false)

// ---------------- degree / utility kernels ----------------

__global__ __launch_bounds__(256) void k_fill1(float* __restrict__ p, int n) {
    int i = blockIdx.x * 256 + threadIdx.x;
    if (i < n) p[i] = 1.0f;  // self-loop contributes 1 to every degree
}

__global__ __launch_bounds__(256) void k_deg_accum(float* __restrict__ deg,
                                                   const int* __restrict__ row, int e0) {
    int i = blockIdx.x * 256 + threadIdx.x;
    if (i < e0) unsafeAtomicAdd(&deg[row[i]], 1.0f);
}

__global__ __launch_bounds__(256) void k_rsqrt(float* __restrict__ p, int n) {
    int i = blockIdx.x * 256 + threadIdx.x;
    if (i < n) p[i] = rsqrtf(p[i]);
}

__global__ __launch_bounds__(256) void k_zero4(float4* __restrict__ p, int n4) {
    int i = blockIdx.x * 256 + threadIdx.x;
    if (i < n4) p[i] = make_float4(0.f, 0.f, 0.f, 0.f);
}

// Copy W (300x300) into zero-padded Wp (304x300): no per-step B masking needed.
__global__ __launch_bounds__(256) void k_padW(const float* __restrict__ W,
                                              float* __restrict__ Wp) {
    int i = blockIdx.x * 256 + threadIdx.x;
    if (i < NPAD * DF) Wp[i] = (i < DF * DF) ? W[i] : 0.f;
}

// ---------------- GEMM: out[m][n] = sum_k A[m][k] * W[n][k] + bias[n] ----------------
// Block = 256 threads (8 waves), one 16-row M tile per block.
// A panel staged in LDS in fragment order: lds[s*FRAG_STRIDE + lane*2 + j]
//   = A[lane&15][4s + 2*(lane>>4) + j]  -> one aligned ds_load_b64 per K-step.
// Each wave owns two N tiles (w, w+8) sharing the A fragment; depth-3 software
// pipeline with *named* slot registers and a branch-free steady-state loop.

__global__ __launch_bounds__(256) void k_gemm(const float* __restrict__ A,
                                              const float* __restrict__ Wp,
                                              const float* __restrict__ bias,
                                              float* __restrict__ out,
                                              int M, int relu_in) {
    __shared__ float lds[K4 * FRAG_STRIDE];  // 75*66*4 = 19.8 KB
    const int m0 = blockIdx.x * 16;
    const int t  = threadIdx.x;

    // Cooperative stage: coalesced float4 global reads, pre-swizzled LDS writes.
    for (int i = t; i < 16 * K4; i += 256) {
        int r  = i / K4;          // row within tile
        int c4 = i - r * K4;      // K-step
        int m  = m0 + r;
        float4 v = make_float4(0.f, 0.f, 0.f, 0.f);
        if (m < M) v = ((const float4*)(A + (size_t)m * DF))[c4];
        if (relu_in) {
            v.x = fmaxf(v.x, 0.f); v.y = fmaxf(v.y, 0.f);
            v.z = fmaxf(v.z, 0.f); v.w = fmaxf(v.w, 0.f);
        }
        *(float2*)&lds[c4 * FRAG_STRIDE + r * 2]      = make_float2(v.x, v.y);
        *(float2*)&lds[c4 * FRAG_STRIDE + r * 2 + 32] = make_float2(v.z, v.w);
    }
    __syncthreads();

    const int wave = t >> 5;
    const int lane = t & 31;
    const int kh   = (lane >> 4) << 1;   // K sub-offset: 0 or 2
    const int ln   = lane & 15;
    const float* af = &lds[lane * 2];    // + s*FRAG_STRIDE per step
    const int mbase = m0 + ((lane >> 4) << 3);

#define LDA(s)    (*(const v2f*)(af + (s) * FRAG_STRIDE))
#define LDW(p, s) (*(const v2f*)((p) + (s) * 4))

    // ---- pass 1: tiles w and w+8 (n <= 255, always valid) ----
    {
        const int n0 = wave * 16 + ln;
        const int n1 = (wave + 8) * 16 + ln;
        const float* w0 = Wp + (size_t)n0 * DF + kh;
        const float* w1 = Wp + (size_t)n1 * DF + kh;

        v8f c0 = {}, c1 = {};
        v2f a0 = LDA(0), a1 = LDA(1), a2 = LDA(2);
        v2f p0 = LDW(w0, 0), p1 = LDW(w0, 1), p2 = LDW(w0, 2);
        v2f q0 = LDW(w1, 0), q1 = LDW(w1, 1), q2 = LDW(w1, 2);

        for (int s = 0; s < K4 - PIPE; s += PIPE) {   // 24 branch-free iterations
            c0 = WMMA_F32(a0, p0, c0);  c1 = WMMA_F32(a0, q0, c1);
            a0 = LDA(s + 3);  p0 = LDW(w0, s + 3);  q0 = LDW(w1, s + 3);
            c0 = WMMA_F32(a1, p1, c0);  c1 = WMMA_F32(a1, q1, c1);
            a1 = LDA(s + 4);  p1 = LDW(w0, s + 4);  q1 = LDW(w1, s + 4);
            c0 = WMMA_F32(a2, p2, c0);  c1 = WMMA_F32(a2, q2, c1);
            a2 = LDA(s + 5);  p2 = LDW(w0, s + 5);  q2 = LDW(w1, s + 5);
        }
        // epilogue: drain the 3 slots
        c0 = WMMA_F32(a0, p0, c0);  c1 = WMMA_F32(a0, q0, c1);
        c0 = WMMA_F32(a1, p1, c0);  c1 = WMMA_F32(a1, q1, c1);
        c0 = WMMA_F32(a2, p2, c0);  c1 = WMMA_F32(a2, q2, c1);

        const float bv0 = bias[n0];
        const float bv1 = bias[n1];
        #pragma unroll
        for (int r = 0; r < 8; ++r) {
            int m = mbase + r;
            if (m < M) {
                out[(size_t)m * DF + n0] = c0[r] + bv0;
                out[(size_t)m * DF + n1] = c1[r] + bv1;
            }
        }
    }

    // ---- pass 2: tiles 16..18 on waves 0..2 (n in 256..303; W rows padded) ----
    if (wave < 3) {
        const int n = (16 + wave) * 16 + ln;
        const float* w0 = Wp + (size_t)n * DF + kh;

        v8f c0 = {};
        v2f a0 = LDA(0), a1 = LDA(1), a2 = LDA(2);
        v2f p0 = LDW(w0, 0), p1 = LDW(w0, 1), p2 = LDW(w0, 2);

        for (int s = 0; s < K4 - PIPE; s += PIPE) {
            c0 = WMMA_F32(a0, p0, c0);
            a0 = LDA(s + 3);  p0 = LDW(w0, s + 3);
            c0 = WMMA_F32(a1, p1, c0);
            a1 = LDA(s + 4);  p1 = LDW(w0, s + 4);
            c0 = WMMA_F32(a2, p2, c0);
            a2 = LDA(s + 5);  p2 = LDW(w0, s + 5);
        }
        c0 = WMMA_F32(a0, p0, c0);
        c0 = WMMA_F32(a1, p1, c0);
        c0 = WMMA_F32(a2, p2, c0);

        if (n < DF) {
            const float bv = bias[n];
            #pragma unroll
            for (int r = 0; r < 8; ++r) {
                int m = mbase + r;
                if (m < M) out[(size_t)m * DF + n] = c0[r] + bv;
            }
        }
    }
#undef LDA
#undef LDW
}

// ---------------- edge scatter: out[col] += dis[row]*dis[col] * h[row] ----------------

__global__ __launch_bounds__(256) void k_scatter(const float* __restrict__ h,
                                                 const int* __restrict__ ei,
                                                 const float* __restrict__ dis,
                                                 float* __restrict__ out,
                                                 int E0, int Etot) {
    const int wv = blockIdx.x * 8 + (threadIdx.x >> 5);
    if (wv >= Etot) return;
    const int lane = threadIdx.x & 31;

    int r, c;
    if (wv < E0) { r = ei[wv]; c = ei[E0 + wv]; }
    else         { r = wv - E0; c = r; }          // appended self loop

    const float norm = dis[r] * dis[c];
    const float4* hr = (const float4*)(h + (size_t)r * DF);
    float* oc = out + (size_t)c * DF;

    for (int q = lane; q < K4; q += 32) {
        float4 v = hr[q];
        int d = q << 2;
        unsafeAtomicAdd(oc + d + 0, norm * v.x);
        unsafeAtomicAdd(oc + d + 1, norm * v.y);
        unsafeAtomicAdd(oc + d + 2, norm * v.z);
        unsafeAtomicAdd(oc + d + 3, norm * v.w);
    }
}

// ---------------- driver ----------------

extern "C" void kernel_launch(void* const* d_in, const int* in_sizes, int n_in,
                              void* d_out, int out_size, void* d_ws, size_t ws_size,
                              hipStream_t stream) {
    const float* x  = (const float*)d_in[0];
    const int*   ei = (const int*)d_in[1];   // [2, E0] int32, row-major
    const float* W1 = (const float*)d_in[2];
    const float* b1 = (const float*)d_in[3];
    const float* W2 = (const float*)d_in[4];
    const float* b2 = (const float*)d_in[5];
    float* out = (float*)d_out;

    const int N    = in_sizes[0] / DF;   // 100000
    const int E0   = in_sizes[1] / 2;    // 800000
    const int Etot = E0 + N;

    // workspace: dis (N) | h (N*DF) | g (N*DF) | Wpad (304*300)
    float* dis = (float*)d_ws;
    size_t disPad = ((size_t)N + 127) & ~(size_t)127;
    float* h  = dis + disPad;
    float* g  = h + (size_t)N * DF;
    float* wp = g + (size_t)N * DF;

    const int mtiles = (N + 15) / 16;
    const int nfeat4 = (N * DF) / 4;
    const int npadW  = (NPAD * DF + 255) / 256;

    // degrees -> deg^-0.5
    k_fill1    <<<(N  + 255) / 256, 256, 0, stream>>>(dis, N);
    k_deg_accum<<<(E0 + 255) / 256, 256, 0, stream>>>(dis, ei, E0);
    k_rsqrt    <<<(N  + 255) / 256, 256, 0, stream>>>(dis, N);

    // layer 1
    k_padW   <<<npadW, 256, 0, stream>>>(W1, wp);
    k_gemm   <<<mtiles, 256, 0, stream>>>(x, wp, b1, h, N, 0);
    k_zero4  <<<(nfeat4 + 255) / 256, 256, 0, stream>>>((float4*)g, nfeat4);
    k_scatter<<<(Etot + 7) / 8, 256, 0, stream>>>(h, ei, dis, g, E0, Etot);

    // layer 2 (ReLU fused into GEMM's A staging)
    k_padW   <<<npadW, 256, 0, stream>>>(W2, wp);
    k_gemm   <<<mtiles, 256, 0, stream>>>(g, wp, b2, h, N, 1);
    k_zero4  <<<(nfeat4 + 255) / 256, 256, 0, stream>>>((float4*)out, nfeat4);
    k_scatter<<<(Etot + 7) / 8, 256, 0, stream>>>(h, ei, dis, out, E0, Etot);
}